// KPlanesFeatureEncoder_71631464563310
// MI455X (gfx1250) — compile-verified
//
#include <hip/hip_runtime.h>

#define N_PLANES 18

struct PlaneDesc { const float* src; float* dst; };
struct PlaneTab  { PlaneDesc d[N_PLANES]; };

// ---- gfx1250 async global->LDS path (guarded; falls back to plain LDS stores) ----
// Probe-verified: builtin exists, params are (as1 int*, as3 int*, imm offset, imm cpol).
#if __has_builtin(__builtin_amdgcn_global_load_async_to_lds_b32)
#define KP_ASYNC_LDS 1
typedef __attribute__((address_space(1))) int kp_glb_i32;
typedef __attribute__((address_space(3))) int kp_lds_i32;
#else
#define KP_ASYNC_LDS 0
#endif

// ----------------------------------------------------------------------------------
// Pre-pass: transpose each plane [C=32, H, W] -> [H, W, C] so the 32 channels of one
// bilinear corner become a single contiguous 128B chunk (one wave32 coalesced load).
// Tiles of 32 pixels x 32 channels staged through LDS; the global->LDS fill uses the
// async copy engine's per-lane LDS scatter (ASYNCcnt), read-back is conflict-free ds.
// ----------------------------------------------------------------------------------
__global__ __launch_bounds__(256) void transpose_chw_hwc(const float* __restrict__ src,
                                                         float* __restrict__ dst,
                                                         int HW) {
  __shared__ float tile[32][33];  // +1 pad: conflict-free column writes
  const int lane = threadIdx.x & 31;   // pixel offset within tile (load phase)
  const int row  = threadIdx.x >> 5;   // 0..7
  const int base = (int)blockIdx.x * 32;
  const int pix  = base + lane;

  if (pix < HW) {
#pragma unroll
    for (int i = 0; i < 4; ++i) {
      const int ch  = row + (i << 3);        // 8 waves x 4 iters = 32 channels
      const int off = ch * HW + pix;         // < 2^31: keep address math 32-bit
#if KP_ASYNC_LDS
      __builtin_amdgcn_global_load_async_to_lds_b32(
          (kp_glb_i32*)(src + off), (kp_lds_i32*)&tile[lane][ch], 0, 0);
#else
      tile[lane][ch] = src[off];
#endif
    }
  }
#if KP_ASYNC_LDS
#if __has_builtin(__builtin_amdgcn_s_wait_asynccnt)
  __builtin_amdgcn_s_wait_asynccnt(0);
#else
  asm volatile("s_wait_asynccnt 0" ::: "memory");
#endif
#endif
  __syncthreads();

#pragma unroll
  for (int i = 0; i < 4; ++i) {
    const int p = row + (i << 3);            // pixel within tile (store phase)
    if (base + p < HW)
      dst[(base + p) * 32 + lane] = tile[p][lane];  // lane = channel, coalesced
  }
}

__device__ __forceinline__ float kp_lerp(float a, float b, float w) {
  return a * (1.0f - w) + b * w;
}

// ----------------------------------------------------------------------------------
// Main encoder: one wave32 per point, lane = feature channel (FEAT_DIM == 32).
// All plane shapes are compile-time constants (fully unrolled) -> shift addressing.
// 72 independent gathers/thread feed the L2; grids (~50MB) stay L2-resident (192MB).
// Outputs (268MB stream) use non-temporal stores to avoid evicting the grids.
// ----------------------------------------------------------------------------------
template <bool HWC>
__global__ __launch_bounds__(256) void kplanes_encode(const float4* __restrict__ xyzt,
                                                      const float*  __restrict__ aabb,
                                                      PlaneTab tab,
                                                      float* __restrict__ out_v3d,
                                                      float* __restrict__ out_v4d,
                                                      int N) {
  constexpr int RESO[3][4] = {{64, 64, 64, 100}, {128, 128, 128, 100}, {256, 256, 256, 100}};
  constexpr int C1[6] = {0, 0, 0, 1, 1, 2};
  constexpr int C2[6] = {1, 2, 3, 2, 3, 3};

  const int tid = (int)blockIdx.x * 256 + (int)threadIdx.x;
  const int n   = tid >> 5;
  const int c   = threadIdx.x & 31;
  if (n >= N) return;

  const float4 p = xyzt[n];  // wave-uniform 16B load
  float pc[4];
  {
    const float in[4] = {p.x, p.y, p.z, p.w};
#pragma unroll
    for (int j = 0; j < 4; ++j) {
      const float lo = aabb[j], hi = aabb[4 + j];
      pc[j] = (in[j] - lo) * (2.0f / (hi - lo)) - 1.0f;  // normalize to [-1,1]
    }
  }

  float v3_0 = 0.0f;
  float v4[3];
#pragma unroll
  for (int s = 0; s < 3; ++s) {
    float f[6];
#pragma unroll
    for (int g = 0; g < 6; ++g) {
      const int   W  = RESO[s][C1[g]];
      const int   H  = RESO[s][C2[g]];
      const float cx = pc[C1[g]];
      const float cy = pc[C2[g]];

      // grid_sample(align_corners=True, padding_mode='border')
      float fx  = fminf(fmaxf((cx + 1.0f) * 0.5f * (float)(W - 1), 0.0f), (float)(W - 1));
      float fy  = fminf(fmaxf((cy + 1.0f) * 0.5f * (float)(H - 1), 0.0f), (float)(H - 1));
      float x0f = floorf(fx), y0f = floorf(fy);
      int   x0  = (int)x0f,   y0  = (int)y0f;
      int   x1  = x0 + 1; if (x1 > W - 1) x1 = W - 1;
      int   y1  = y0 + 1; if (y1 > H - 1) y1 = H - 1;
      const float wx = fx - x0f, wy = fy - y0f;

      const int pidx = s * 6 + g;
      float p00, p01, p10, p11;
      if (HWC) {
        // [H, W, C]: channels contiguous -> each corner is one coalesced 128B wave load
        const float* pl = tab.d[pidx].dst;
        const int r0 = (y0 * W) * 32 + c;    // all intra-plane offsets fit in 32-bit
        const int r1 = (y1 * W) * 32 + c;
        p00 = pl[r0 + x0 * 32]; p01 = pl[r0 + x1 * 32];
        p10 = pl[r1 + x0 * 32]; p11 = pl[r1 + x1 * 32];
      } else {
        // fallback: original [C, H, W] layout (channel-strided gathers)
        const float* pl = tab.d[pidx].src;
        const int cb = c * (W * H);
        p00 = pl[cb + y0 * W + x0]; p01 = pl[cb + y0 * W + x1];
        p10 = pl[cb + y1 * W + x0]; p11 = pl[cb + y1 * W + x1];
      }
      f[g] = kp_lerp(kp_lerp(p00, p01, wx), kp_lerp(p10, p11, wx), wy);
    }
    const float v3  = f[0] * f[1] * f[3];          // f_xy * f_xz * f_yz
    const float vv4 = v3 * f[2] * f[4] * f[5];     // * f_xt * f_yt * f_zt
    if (s == 0) v3_0 = v3;
    v4[s] = vv4;
  }

  // streaming outputs: non-temporal so the 268MB result doesn't evict L2-resident grids
  __builtin_nontemporal_store(v3_0, out_v3d + (size_t)n * 32 + c);
#pragma unroll
  for (int s = 0; s < 3; ++s)
    __builtin_nontemporal_store(v4[s], out_v4d + (size_t)n * 96 + s * 32 + c);
}

extern "C" void kernel_launch(void* const* d_in, const int* in_sizes, int n_in,
                              void* d_out, int out_size, void* d_ws, size_t ws_size,
                              hipStream_t stream) {
  static const int RESO[3][4] = {{64, 64, 64, 100}, {128, 128, 128, 100}, {256, 256, 256, 100}};
  static const int C1[6] = {0, 0, 0, 1, 1, 2};
  static const int C2[6] = {1, 2, 3, 2, 3, 3};

  const float* xyzt = (const float*)d_in[0];
  const float* aabb = (const float*)d_in[1];
  const int N = in_sizes[0] / 4;

  // Plane table: inputs are xyzt, aabb, then 18 planes (scale-major, COO order);
  // also handle the case where the whole grids tuple arrives as one flat buffer.
  PlaneTab tab;
  size_t wsFloats = 0;
  size_t catOff = 0;
  const float* concat = (n_in >= 2 + N_PLANES) ? nullptr : (const float*)d_in[2];
  for (int s = 0; s < 3; ++s) {
    for (int g = 0; g < 6; ++g) {
      const int p = s * 6 + g;
      const int W = RESO[s][C1[g]];
      const int H = RESO[s][C2[g]];
      const size_t sz = (size_t)32 * W * H;
      tab.d[p].src = concat ? (concat + catOff) : (const float*)d_in[2 + p];
      tab.d[p].dst = (float*)d_ws + wsFloats;
      catOff  += sz;
      wsFloats += sz;
    }
  }

  const bool useHWC = (ws_size >= wsFloats * sizeof(float));
  if (useHWC) {
    for (int p = 0; p < N_PLANES; ++p) {
      const int g = p % 6, s = p / 6;
      const int HW = RESO[s][C1[g]] * RESO[s][C2[g]];
      const int tiles = (HW + 31) / 32;
      transpose_chw_hwc<<<tiles, 256, 0, stream>>>(tab.d[p].src, tab.d[p].dst, HW);
    }
  }

  float* out_v3d = (float*)d_out;
  float* out_v4d = (float*)d_out + (size_t)N * 32;
  const int blocks = (N + 7) / 8;  // 8 points (waves) per 256-thread block
  if (useHWC)
    kplanes_encode<true><<<blocks, 256, 0, stream>>>(
        (const float4*)xyzt, aabb, tab, out_v3d, out_v4d, N);
  else
    kplanes_encode<false><<<blocks, 256, 0, stream>>>(
        (const float4*)xyzt, aabb, tab, out_v3d, out_v4d, N);
}